// PaCoVoxelLoss_56307021250584
// MI455X (gfx1250) — compile-verified
//
#include <hip/hip_runtime.h>
#include <math.h>

typedef __attribute__((ext_vector_type(2))) float v2f;
typedef __attribute__((ext_vector_type(8))) float v8f;

#define Bn 8
#define Qn 20
#define Pn 12
#define Nn 4096
#define Mn 1280      // NUM_QUERIES * FACTOR
#define Gn 2048
#define FACTORn 64
#define KNB 8

// ---- workspace layout (units: 4-byte words) ----
enum {
  WS_COST = 0,                   // B*Q*P floats
  WS_CH   = WS_COST + Bn*Qn*Pn,  // cham_raw
  WS_NS   = WS_CH   + Bn*Qn*Pn,  // normal_sim
  WS_DD   = WS_NS   + Bn*Qn*Pn,  // dist_diff
  WS_HAS  = WS_DD   + Bn*Qn*Pn,  // B*P floats (0/1)
  WS_ROWS = WS_HAS  + Bn*Pn,     // B*P ints
  WS_SCAL = WS_ROWS + Bn*Pn,     // [cls, param, pvd, rep, ...] floats
  WS_D1M  = WS_SCAL + 8,         // B*M uint bits of running min (L1, over g)
  WS_D2M  = WS_D1M  + Bn*Mn,     // B*M (L2)
  WS_D1G  = WS_D2M  + Bn*Mn,     // B*G (L1, over m)
  WS_D2G  = WS_D1G  + Bn*Gn,     // B*G (L2)
  WS_END  = WS_D2G  + Bn*Gn
};

// ---------------------------------------------------------------- init
__global__ void k_init(unsigned int* __restrict__ wsu) {
  int i = blockIdx.x * blockDim.x + threadIdx.x;
  if (i < 8) wsu[WS_SCAL + i] = 0u;                  // 0.0f
  int nmin = 2 * Bn * Mn + 2 * Bn * Gn;              // contiguous min arrays
  if (i < nmin) wsu[WS_D1M + i] = 0x7f800000u;       // +inf
}

// ------------------------------------------------------- cost matrix
// pp[q][n] = |points . n_q - d_q| ; cham[q][p] = sum_n pp*mask / max(cnt,1)
__global__ void k_cost(const float* __restrict__ pred_normals,
                       const float* __restrict__ pred_dist,
                       const float* __restrict__ gt_normals,
                       const float* __restrict__ gt_dist,
                       const unsigned char* __restrict__ gt_masks,
                       const float* __restrict__ points,
                       float* __restrict__ ws) {
  __shared__ float pn[Qn * 3], pd[Qn], gn[Pn * 3], gd[Pn];
  __shared__ float cham[Qn * Pn];
  __shared__ float cnt[Pn];
  int b = blockIdx.x, t = threadIdx.x;
  if (t < Qn * 3) pn[t] = pred_normals[b * Qn * 3 + t];
  if (t < Qn)     pd[t] = pred_dist[b * Qn + t];
  if (t < Pn * 3) gn[t] = gt_normals[b * Pn * 3 + t];
  if (t < Pn)     gd[t] = gt_dist[b * Pn + t];
  if (t < Qn * Pn) cham[t] = 0.f;
  if (t < Pn)      cnt[t] = 0.f;
  __syncthreads();
  for (int n = t; n < Nn; n += blockDim.x) {
    float px = points[(b * Nn + n) * 3 + 0];
    float py = points[(b * Nn + n) * 3 + 1];
    float pz = points[(b * Nn + n) * 3 + 2];
    float pp[Qn];
#pragma unroll
    for (int q = 0; q < Qn; q++)
      pp[q] = fabsf(px * pn[q * 3] + py * pn[q * 3 + 1] + pz * pn[q * 3 + 2] - pd[q]);
    for (int p = 0; p < Pn; p++) {
      if (gt_masks[(b * Pn + p) * Nn + n]) {
        atomicAdd(&cnt[p], 1.0f);
#pragma unroll
        for (int q = 0; q < Qn; q++) atomicAdd(&cham[q * Pn + p], pp[q]);
      }
    }
  }
  __syncthreads();
  if (t < Qn * Pn) {
    int q = t / Pn, p = t % Pn;
    float c  = cnt[p];
    float ch = cham[t] / fmaxf(c, 1.f);
    float ns = 1.f - fabsf(pn[q * 3] * gn[p * 3] + pn[q * 3 + 1] * gn[p * 3 + 1] +
                           pn[q * 3 + 2] * gn[p * 3 + 2]);
    float dd = fabsf(pd[q] - gd[p]);
    float cost = ns + 0.5f * dd + 5.f * ((c > 0.f) ? ch : 1.f);
    int o = b * Qn * Pn + t;
    ws[WS_COST + o] = cost;
    ws[WS_CH + o]   = ch;
    ws[WS_NS + o]   = ns;
    ws[WS_DD + o]   = dd;
  }
  if (t < Pn) ws[WS_HAS + b * Pn + t] = (cnt[t] > 0.f) ? 1.f : 0.f;
}

// ----------------------------------------------- Hungarian (JV) per batch
__global__ void k_hungarian(float* __restrict__ ws) {
  int b = threadIdx.x;
  if (b >= Bn) return;
  const int n = Qn;
  double sq[Qn][Qn];
  for (int q = 0; q < Qn; q++)
    for (int j = 0; j < Qn; j++)
      sq[q][j] = (j < Pn) ? (double)ws[WS_COST + b * Qn * Pn + q * Pn + j] : 1e6;
  double u[Qn + 1], v[Qn + 1];
  int p[Qn + 1], way[Qn + 1];
  for (int j = 0; j <= n; j++) { u[j] = 0; v[j] = 0; p[j] = 0; way[j] = 0; }
  for (int i = 1; i <= n; i++) {
    p[0] = i; int j0 = 0;
    double minv[Qn + 1]; bool used[Qn + 1];
    for (int j = 0; j <= n; j++) { minv[j] = 1e18; used[j] = false; }
    while (true) {
      used[j0] = true; int i0 = p[j0]; double delta = 1e18; int j1 = 0;
      for (int j = 1; j <= n; j++) if (!used[j]) {
        double cur = sq[i0 - 1][j - 1] - u[i0] - v[j];
        if (cur < minv[j]) { minv[j] = cur; way[j] = j0; }
        if (minv[j] < delta) { delta = minv[j]; j1 = j; }
      }
      for (int j = 0; j <= n; j++) {
        if (used[j]) { u[p[j]] += delta; v[j] -= delta; }
        else minv[j] -= delta;
      }
      j0 = j1;
      if (p[j0] == 0) break;
    }
    while (j0 != 0) { int j1 = way[j0]; p[j0] = p[j1]; j0 = j1; }
  }
  int* rows = (int*)ws + WS_ROWS;
  for (int j = 0; j < Pn; j++) rows[b * Pn + j] = p[j + 1] - 1;
}

// ------------------------------------------ cls / param / pvd losses
__global__ void k_small(const float* __restrict__ pred_logits, float* __restrict__ ws) {
  __shared__ float s_cls, s_param, s_pvd;
  int t = threadIdx.x;
  if (t == 0) { s_cls = 0.f; s_param = 0.f; s_pvd = 0.f; }
  __syncthreads();
  const int* rows = (const int*)ws + WS_ROWS;
  if (t < Bn * Qn) {
    int b = t / Qn, q = t % Qn;
    float tgt = 0.f;
    for (int p = 0; p < Pn; p++) if (rows[b * Pn + p] == q) tgt += 1.f;
    float x = pred_logits[t];
    float bce = fmaxf(x, 0.f) - x * tgt + log1pf(expf(-fabsf(x)));
    atomicAdd(&s_cls, bce);
  }
  if (t < Bn * Pn) {
    int b = t / Pn, p = t % Pn;
    int r = rows[b * Pn + p];
    int o = b * Qn * Pn + r * Pn + p;
    atomicAdd(&s_param, ws[WS_NS + o] + ws[WS_DD + o]);
    atomicAdd(&s_pvd, (ws[WS_HAS + b * Pn + p] > 0.f) ? ws[WS_CH + o] : 0.f);
  }
  __syncthreads();
  if (t == 0) {
    ws[WS_SCAL + 0] = s_cls / (float)(Bn * Qn);
    ws[WS_SCAL + 1] = s_param / (float)(Bn * Pn);
    ws[WS_SCAL + 2] = s_pvd / (float)(Bn * Pn);
  }
}

// -------------------------------------------------- repulsion (160 groups)
__global__ void k_repulsion(const float* __restrict__ rp, float* __restrict__ ws) {
  __shared__ float pts[FACTORn * 3];
  __shared__ float red[FACTORn];
  int g = blockIdx.x;           // b*Q + q
  int f = threadIdx.x;
  int base = g * FACTORn * 3;
  pts[f * 3 + 0] = rp[base + f * 3 + 0];
  pts[f * 3 + 1] = rp[base + f * 3 + 1];
  pts[f * 3 + 2] = rp[base + f * 3 + 2];
  __syncthreads();
  float x0 = pts[f * 3], x1 = pts[f * 3 + 1], x2 = pts[f * 3 + 2];
  float best[KNB];
#pragma unroll
  for (int j = 0; j < KNB; j++) best[j] = 1e30f;
  for (int j = 0; j < FACTORn; j++) {
    float dx = x0 - pts[j * 3], dy = x1 - pts[j * 3 + 1], dz = x2 - pts[j * 3 + 2];
    float d2 = dx * dx + dy * dy + dz * dz;
    if (d2 < best[KNB - 1]) {          // keep 8 smallest, sorted ascending
      best[KNB - 1] = d2;
#pragma unroll
      for (int k = KNB - 1; k > 0; k--)
        if (best[k] < best[k - 1]) { float tmp = best[k]; best[k] = best[k - 1]; best[k - 1] = tmp; }
    }
  }
  float s = 0.f;
#pragma unroll
  for (int j = 1; j < KNB; j++) {       // skip self (best[0] == 0)
    float dn = fmaxf(best[j], 1e-12f);
    float w  = expf(-dn * (1.f / (0.03f * 0.03f)));
    s += (0.07f - sqrtf(dn)) * w;
  }
  red[f] = s;
  __syncthreads();
  for (int st = FACTORn / 2; st > 0; st >>= 1) {
    if (f < st) red[f] += red[f + st];
    __syncthreads();
  }
  if (f == 0) {
    float meanv = red[0] / (float)(FACTORn * (KNB - 1));
    atomicAdd(&ws[WS_SCAL + 3], fmaxf(meanv, 0.f) / (float)(Bn * Qn));
  }
}

// ------------------------------- fine Chamfer: WMMA f32 16x16x4 for L2 tiles
// d2[m][g] = [x0,x1,x2,|x|^2] . [-2y0,-2y1,-2y2,1] + |y|^2  (exact, K=4)
__global__ void k_chamfer(const float* __restrict__ recon, const float* __restrict__ gt,
                          unsigned int* __restrict__ wsu) {
  __shared__ float xt[16 * 3];
  __shared__ float xsq[16];
  int b  = blockIdx.x / (Mn / 16);
  int mt = blockIdx.x % (Mn / 16);
  int t  = threadIdx.x;
  if (t < 16) {
    int m = mt * 16 + t;
    float a = recon[(b * Mn + m) * 3 + 0];
    float c = recon[(b * Mn + m) * 3 + 1];
    float e = recon[(b * Mn + m) * 3 + 2];
    xt[t * 3] = a; xt[t * 3 + 1] = c; xt[t * 3 + 2] = e;
    xsq[t] = a * a + c * c + e * e;
  }
  __syncthreads();
  int w = t >> 5, lane = t & 31;
  int half = lane >> 4, mi = lane & 15;
  // A fragment (16x4 f32): lanes 0-15 -> K0/K1, lanes 16-31 -> K2/K3
  v2f A;
  A.x = half ? xt[mi * 3 + 2] : xt[mi * 3 + 0];
  A.y = half ? xsq[mi]        : xt[mi * 3 + 1];
  float rm1[8], rm2[8];
#pragma unroll
  for (int r = 0; r < 8; r++) { rm1[r] = 1e30f; rm2[r] = 1e30f; }
  unsigned int* d1g = wsu + WS_D1G;
  unsigned int* d2g = wsu + WS_D2G;
  for (int tg = w; tg < Gn / 16; tg += 8) {   // uniform per wave: EXEC all-ones
    int gidx = tg * 16 + mi;
    float y0 = gt[(b * Gn + gidx) * 3 + 0];
    float y1 = gt[(b * Gn + gidx) * 3 + 1];
    float y2 = gt[(b * Gn + gidx) * 3 + 2];
    float ysq = y0 * y0 + y1 * y1 + y2 * y2;
    // B fragment (4x16 f32): lanes 0-15 -> K0/K1, lanes 16-31 -> K2/K3
    v2f Bm;
    Bm.x = half ? (-2.f * y2) : (-2.f * y0);
    Bm.y = half ? 1.0f        : (-2.f * y1);
    v8f C = {ysq, ysq, ysq, ysq, ysq, ysq, ysq, ysq};
    v8f D = __builtin_amdgcn_wmma_f32_16x16x4_f32(false, A, false, Bm, (short)0, C,
                                                  false, false);
    float g1min = 1e30f, g2min = 1e30f;
#pragma unroll
    for (int r = 0; r < 8; r++) {
      int mrow = r + 8 * half;                // D layout: VGPR r, lane l -> M=r+8*(l>=16)
      float l1 = fabsf(xt[mrow * 3 + 0] - y0) + fabsf(xt[mrow * 3 + 1] - y1) +
                 fabsf(xt[mrow * 3 + 2] - y2);
      float l2 = D[r];
      rm1[r] = fminf(rm1[r], l1);
      rm2[r] = fminf(rm2[r], l2);
      g1min = fminf(g1min, l1);
      g2min = fminf(g2min, l2);
    }
    // min over m for this g (bit-compare valid: distances >= 0)
    atomicMin(&d1g[b * Gn + gidx], __float_as_uint(g1min));
    atomicMin(&d2g[b * Gn + gidx], __float_as_uint(g2min));
  }
  unsigned int* d1m = wsu + WS_D1M;
  unsigned int* d2m = wsu + WS_D2M;
#pragma unroll
  for (int r = 0; r < 8; r++) {
    int m = mt * 16 + r + 8 * half;
    atomicMin(&d1m[b * Mn + m], __float_as_uint(rm1[r]));
    atomicMin(&d2m[b * Mn + m], __float_as_uint(rm2[r]));
  }
}

// ---------------------------------------------------------- final combine
__global__ void k_finish(const unsigned int* __restrict__ wsu,
                         const float* __restrict__ wsf, float* __restrict__ out) {
  __shared__ float s1m[Bn], s2m[Bn], s1g[Bn], s2g[Bn];
  int t = threadIdx.x;
  if (t < Bn) { s1m[t] = 0.f; s2m[t] = 0.f; s1g[t] = 0.f; s2g[t] = 0.f; }
  __syncthreads();
  for (int i = t; i < Bn * Mn; i += blockDim.x) {
    int b = i / Mn;
    atomicAdd(&s1m[b], __uint_as_float(wsu[WS_D1M + i]));
    atomicAdd(&s2m[b], __uint_as_float(wsu[WS_D2M + i]));
  }
  for (int i = t; i < Bn * Gn; i += blockDim.x) {
    int b = i / Gn;
    atomicAdd(&s1g[b], __uint_as_float(wsu[WS_D1G + i]));
    atomicAdd(&s2g[b], __uint_as_float(wsu[WS_D2G + i]));
  }
  __syncthreads();
  if (t == 0) {
    float ch1 = 0.f, ch2 = 0.f;
    for (int b = 0; b < Bn; b++) {
      ch1 += 0.5f * (s1m[b] / (float)Mn + s1g[b] / (float)Gn);
      ch2 += 0.5f * (s2m[b] / (float)Mn + s2g[b] / (float)Gn);
    }
    ch1 /= (float)Bn; ch2 /= (float)Bn;
    // total = W_CLS*cls + ALPHA*param + BETA*pvd + W_REP*rep + ch1 + ch2
    out[0] = wsf[WS_SCAL + 0] + 0.5f * wsf[WS_SCAL + 1] + 20.f * wsf[WS_SCAL + 2] +
             wsf[WS_SCAL + 3] + ch1 + ch2;
  }
}

extern "C" void kernel_launch(void* const* d_in, const int* in_sizes, int n_in,
                              void* d_out, int out_size, void* d_ws, size_t ws_size,
                              hipStream_t stream) {
  const float* pred_logits        = (const float*)d_in[0];
  const float* pred_normals       = (const float*)d_in[1];
  const float* pred_dist          = (const float*)d_in[2];
  const float* gt_normals         = (const float*)d_in[3];
  const float* gt_dist            = (const float*)d_in[4];
  const unsigned char* gt_masks   = (const unsigned char*)d_in[5]; // bool array
  const float* points             = (const float*)d_in[6];
  const float* recon              = (const float*)d_in[7];
  const float* gt                 = (const float*)d_in[8];
  // d_in[9] = gt_index: unused by the reference loss
  float* ws         = (float*)d_ws;
  unsigned int* wsu = (unsigned int*)d_ws;
  float* out        = (float*)d_out;

  int nmin = 2 * Bn * Mn + 2 * Bn * Gn;
  k_init<<<(nmin + 255) / 256, 256, 0, stream>>>(wsu);
  k_cost<<<Bn, 256, 0, stream>>>(pred_normals, pred_dist, gt_normals, gt_dist,
                                 gt_masks, points, ws);
  k_hungarian<<<1, 32, 0, stream>>>(ws);
  k_small<<<1, 256, 0, stream>>>(pred_logits, ws);
  k_repulsion<<<Bn * Qn, FACTORn, 0, stream>>>(recon, ws);
  k_chamfer<<<Bn * (Mn / 16), 256, 0, stream>>>(recon, gt, wsu);
  k_finish<<<1, 256, 0, stream>>>(wsu, ws, out);
}